// WaveletPooling2D_9088150799036
// MI455X (gfx1250) — compile-verified
//
#include <hip/hip_runtime.h>
#include <cstdint>

typedef unsigned int u32;
typedef unsigned long long u64;
typedef __attribute__((ext_vector_type(4))) u32   u32x4;
typedef __attribute__((ext_vector_type(8))) u32   u32x8;
typedef __attribute__((ext_vector_type(4))) float f32x4;

// Problem constants (from reference: (32, 512, 512, 32) f32, NHWC)
constexpr int NB  = 32;
constexpr int H   = 512;
constexpr int W   = 512;
constexpr int C   = 32;
constexpr int OH  = H / 2;           // 256
constexpr int OW  = W / 2;           // 256
constexpr int TW  = 64;              // output pixels per tile
constexpr int TPR = OW / TW;         // 4 tiles per output row
constexpr int IN_TILE_PIX = 2 * TW;  // 128 input pixels per row segment
constexpr int ROW_FLOATS  = IN_TILE_PIX * C;   // 4096 floats = 16 KB per row
constexpr int LDS_FLOATS  = 2 * ROW_FLOATS;    // 8192 floats = 32 KB

__global__ __launch_bounds__(256)
void haar_ll_tdm_kernel(const float* __restrict__ in, float* __restrict__ out) {
    __shared__ float smem[LDS_FLOATS];

    const u32 t    = blockIdx.x;
    const u32 tx   = t & (TPR - 1);          // tile in row
    const u32 rest = t >> 2;                 // TPR == 4
    const u32 oy   = rest & (OH - 1);
    const u32 bb   = rest >> 8;              // OH == 256

    // Global byte address of tile start: (bb, 2*oy, tx*IN_TILE_PIX, 0)
    const u64 elem_off = ((u64)(bb * H + 2 * oy) * (u64)W + (u64)tx * IN_TILE_PIX) * (u64)C;
    const u64 gaddr    = (u64)(uintptr_t)in + elem_off * 4ull;

    if (threadIdx.x < 32) {  // wave 0 only: TDM issues once per wave
        const u32 lds_off = (u32)(uintptr_t)(&smem[0]);  // low 32 bits = LDS byte offset

        // ---- D# group 0 (128b): count=1, lds_addr, global_addr[56:0], type=2 ----
        u32x4 g0;
        g0[0] = 1u;                                            // count=1, user mode
        g0[1] = lds_off;                                       // lds_addr (bytes)
        g0[2] = (u32)gaddr;                                    // global_addr[31:0]
        g0[3] = ((u32)(gaddr >> 32) & 0x01FFFFFFu)             // global_addr[56:32]
              | 0x80000000u;                                   // type=2 at bits[127:126]

        // ---- D# group 1 (256b) ----
        u32x8 g1;
        g1[0] = (2u << 16);                                    // data_size=2 (4 bytes); wg_mask=0
        g1[1] = ((u32)ROW_FLOATS & 0xFFFFu) << 16;             // tensor_dim0[15:0] @ bits[63:48]
        g1[2] = ((u32)ROW_FLOATS >> 16)                        // tensor_dim0[31:16]
              | (2u << 16);                                    // tensor_dim1 = 2 rows (low 16)
        g1[3] = ((u32)ROW_FLOATS & 0xFFFFu) << 16;             // tile_dim0 = 4096 @ bits[127:112]
        g1[4] = 2u;                                            // tile_dim1 = 2 rows; tile_dim2 = 0
        g1[5] = (u32)(W * C);                                  // tensor_dim0_stride = 16384 (low 32)
        g1[6] = 0u;                                            // stride0 hi / stride1 lo
        g1[7] = 0u;                                            // stride1 hi

        u32x4 gz = {0u, 0u, 0u, 0u};                           // zeroed groups 2/3 (3D+ unused)

        asm volatile("tensor_load_to_lds %0, %1, %2, %3"
                     :
                     : "s"(g0), "s"(g1), "s"(gz), "s"(gz)
                     : "memory");
        __builtin_amdgcn_s_wait_tensorcnt(0);
    }
    __syncthreads();

    // Output tile base: (bb, oy, tx*TW, 0)
    f32x4* __restrict__ outv =
        (f32x4*)(out + ((u64)(bb * OH + oy) * (u64)OW + (u64)tx * TW) * (u64)C);

    const u32 tid = threadIdx.x;
#pragma unroll
    for (int k = 0; k < (TW * C / 4) / 256; ++k) {   // 2 iterations
        const u32 v  = tid + (u32)k * 256u;          // output float4 index in tile
        const u32 j  = v >> 3;                       // output pixel in tile (8 vec4 per pixel)
        const u32 cg = v & 7u;                       // channel group
        const u32 f0 = j * (2u * C) + cg * 4u;       // float offset of (row0, pixel 2j, cg)

        const f32x4 a = *(const f32x4*)&smem[f0];                       // (2i,   2j  )
        const f32x4 b = *(const f32x4*)&smem[f0 + C];                   // (2i,   2j+1)
        const f32x4 c = *(const f32x4*)&smem[f0 + ROW_FLOATS];          // (2i+1, 2j  )
        const f32x4 d = *(const f32x4*)&smem[f0 + ROW_FLOATS + C];      // (2i+1, 2j+1)

        f32x4 r = (a + b + c + d) * 0.5f;
        __builtin_nontemporal_store(r, &outv[v]);
    }
}

extern "C" void kernel_launch(void* const* d_in, const int* in_sizes, int n_in,
                              void* d_out, int out_size, void* d_ws, size_t ws_size,
                              hipStream_t stream) {
    const float* in = (const float*)d_in[0];
    float* out      = (float*)d_out;

    const int grid = NB * OH * TPR;  // 32 * 256 * 4 = 32768 workgroups
    haar_ll_tdm_kernel<<<grid, 256, 0, stream>>>(in, out);
}